// Solver_5282809774858
// MI455X (gfx1250) — compile-verified
//
#include <hip/hip_runtime.h>
#include <cstdint>
#include <cstddef>

// ---------------------------------------------------------------------------
// Problem constants (match reference)
// ---------------------------------------------------------------------------
constexpr int   R64 = 64;
constexpr int   NXg = 16384;
constexpr int   NVg = 2048;
#define DXc (2.0f/16384.0f)
#define DVc (16.0f/2048.0f)
#define DTc 0.001f
#define CDIV(a,b) (((a)+(b)-1)/(b))

typedef __attribute__((ext_vector_type(2))) float v2f;
typedef __attribute__((ext_vector_type(8))) float v8f;
typedef __attribute__((ext_vector_type(4))) unsigned u32x4;
typedef __attribute__((ext_vector_type(8))) unsigned u32x8;

// velocity grid value
static __device__ inline float vsf(int k) { return (k + 0.5f) * DVc - 8.0f; }

static __device__ inline float minmodf(float a, float b) {
  return (a * b > 0.f) ? (fabsf(a) < fabsf(b) ? a : b) : 0.f;
}

// ---------------------------------------------------------------------------
// CDNA5 fp32 WMMA: D(16x16) += A(16x4) x B(4x16)
// ---------------------------------------------------------------------------
static __device__ inline v8f wmma4(v2f a, v2f b, v8f c) {
#if __has_builtin(__builtin_amdgcn_wmma_f32_16x16x4_f32)
  return __builtin_amdgcn_wmma_f32_16x16x4_f32(false, a, false, b, (short)0, c,
                                               false, false);
#else
  c[0] += a[0] * b[0];
  return c;
#endif
}

// ---------------------------------------------------------------------------
// gemm64: C[64 x N] = alpha * A[64x64] @ B[64 x N] + beta * C
// 128 threads = 4 waves. A staged via LDS stores; the 64x64 B tile is fetched
// by the Tensor Data Mover (tensor_load_to_lds, 2D descriptor, wave 0 issues,
// completion via s_wait_tensorcnt).  Bs is unpadded: TDM writes contiguous
// rows; fragment reads are lane-stride-1 so no bank conflicts.
// ---------------------------------------------------------------------------
__global__ __launch_bounds__(128)
void gemm64_kernel(const float* __restrict__ A, const float* __restrict__ B,
                   float* __restrict__ C, int N, float alpha, float beta) {
  __shared__ float As[64][65];
  __shared__ float Bs[64][64];
  const int tid  = threadIdx.x;
  const int wave = tid >> 5;
  const int lane = tid & 31;
  const int cb   = blockIdx.x * 64;

  for (int i = tid; i < 4096; i += 128) As[i >> 6][i & 63] = A[i];

  if (tid < 32) {  // wave 0 issues one TDM 2D tile load: B[0:64, cb:cb+64] -> Bs
    unsigned long long ga = (unsigned long long)(uintptr_t)(B + cb);  // byte addr
    unsigned lds = (unsigned)(uintptr_t)&Bs[0][0];
    unsigned nN = (unsigned)N;
    u32x4 g0;
    g0[0] = 1u;                                   // count=1, user mode
    g0[1] = lds;                                  // lds_addr
    g0[2] = (unsigned)(ga & 0xffffffffu);         // global_addr[31:0]
    g0[3] = (unsigned)((ga >> 32) & 0x1ffffffu) | 0x80000000u;  // addr[56:32]|type=2
    u32x8 g1;
    g1[0] = 0x00020000u;                          // wg_mask=0, data_size=2 (4B)
    g1[1] = (nN & 0xffffu) << 16;                 // tensor_dim0[15:0]
    g1[2] = ((nN >> 16) & 0xffffu) | (64u << 16); // tensor_dim0[31:16] | tensor_dim1 lo
    g1[3] = (64u << 16);                          // tensor_dim1 hi=0 | tile_dim0=64
    g1[4] = 64u;                                  // tile_dim1=64, tile_dim2=0
    g1[5] = nN;                                   // tensor_dim0_stride[31:0]
    g1[6] = 0u;                                   // stride hi, dim1_stride lo
    g1[7] = 0u;
    asm volatile("tensor_load_to_lds %0, %1" :: "s"(g0), "s"(g1) : "memory");
    __builtin_amdgcn_s_wait_tensorcnt(0);
  }
  // prefetch the next block's B tile into GL2 (global_prefetch_b8)
  if (cb + 64 < N && tid < 64)
    __builtin_prefetch(&B[(size_t)tid * N + cb + 64], 0, 0);
  __syncthreads();

  const int mr = wave * 16;
  const int lm = lane & 15;
  const int lh = lane >> 4;

  v8f acc[4] = {};
  for (int kk = 0; kk < 16; ++kk) {
    const int k0 = kk * 4 + lh * 2;
    v2f a;
    a.x = As[mr + lm][k0];
    a.y = As[mr + lm][k0 + 1];
#pragma unroll
    for (int nb = 0; nb < 4; ++nb) {
      v2f b;
      b.x = Bs[k0][nb * 16 + lm];
      b.y = Bs[k0 + 1][nb * 16 + lm];
      acc[nb] = wmma4(a, b, acc[nb]);
    }
  }

#pragma unroll
  for (int nb = 0; nb < 4; ++nb)
#pragma unroll
    for (int j = 0; j < 8; ++j) {
      int row = mr + lh * 8 + j;
      int col = cb + nb * 16 + lm;
      if (col < N) {   // tail columns hold TDM overscan; never stored
        size_t idx = (size_t)row * N + col;
        float r = alpha * acc[nb][j];
        if (beta != 0.f) r += beta * C[idx];
        C[idx] = r;
      }
    }
}

// ---------------------------------------------------------------------------
// gram64: part[blk] = (A .* w(k)) @ B^T over this block's K-chunk
// grid = Kd/1024 blocks x 512 threads (16 waves, one 16x16 tile each).
// 64x64 slabs of A and B staged with global_load_async_to_lds_b32 (ASYNCcnt),
// then WMMA from LDS. Deterministic: per-block partials, reduced separately.
// weight modes: 0:1  1:max(vs,0) 2:min(vs,0) 3:vs<0 4:vs>0 5:max(wv,0) 6:min(wv,0)
// ---------------------------------------------------------------------------
static __device__ inline float gram_w(int k, int mode, const float* wv) {
  switch (mode) {
    case 1: { float v = vsf(k); return v > 0.f ? v : 0.f; }
    case 2: { float v = vsf(k); return v < 0.f ? v : 0.f; }
    case 3: { return vsf(k) < 0.f ? 1.f : 0.f; }
    case 4: { return vsf(k) > 0.f ? 1.f : 0.f; }
    case 5: { float e = wv[k]; return e > 0.f ? e : 0.f; }
    case 6: { float e = wv[k]; return e < 0.f ? e : 0.f; }
    default: return 1.f;
  }
}

__global__ __launch_bounds__(512)
void gram64_kernel(const float* __restrict__ A, const float* __restrict__ B,
                   int Kd, float* __restrict__ part, int mode,
                   const float* __restrict__ wv) {
  __shared__ float As[64][65];
  __shared__ float Bs[64][65];
  const int tid  = threadIdx.x;
  const int wave = tid >> 5;
  const int lane = tid & 31;
  const int ti = wave >> 2, tj = wave & 3;
  const int lm = lane & 15, lh = lane >> 4;
  const int chunk = Kd / gridDim.x;
  const int kbeg = blockIdx.x * chunk;
  const int kend = kbeg + chunk;

  v8f acc = {};
  for (int k0 = kbeg; k0 < kend; k0 += 64) {
    __syncthreads();
    // async-stage both 64x64 slabs straight into LDS (ASYNCcnt path)
    for (int i = tid; i < 4096; i += 512) {
      int r = i >> 6, kk = i & 63;
      unsigned la = (unsigned)(uintptr_t)&As[r][kk];
      unsigned lb = (unsigned)(uintptr_t)&Bs[r][kk];
      const float* ga = A + (size_t)r * Kd + k0 + kk;
      const float* gb = B + (size_t)r * Kd + k0 + kk;
      asm volatile("global_load_async_to_lds_b32 %0, %1, off"
                   :: "v"(la), "v"(ga) : "memory");
      asm volatile("global_load_async_to_lds_b32 %0, %1, off"
                   :: "v"(lb), "v"(gb) : "memory");
    }
    asm volatile("s_wait_asynccnt 0" ::: "memory");
    __syncthreads();

    for (int kk = 0; kk < 64; kk += 4) {
      const int ks = kk + lh * 2;
      const int kg = k0 + ks;
      v2f a, b;
      a.x = As[ti * 16 + lm][ks]     * gram_w(kg,     mode, wv);
      a.y = As[ti * 16 + lm][ks + 1] * gram_w(kg + 1, mode, wv);
      b.x = Bs[tj * 16 + lm][ks];
      b.y = Bs[tj * 16 + lm][ks + 1];
      acc = wmma4(a, b, acc);
    }
  }
#pragma unroll
  for (int jj = 0; jj < 8; ++jj) {
    int row = ti * 16 + lh * 8 + jj;
    int col = tj * 16 + lm;
    part[(size_t)blockIdx.x * 4096 + row * 64 + col] = acc[jj];
  }
}

__global__ void gram_reduce_kernel(const float* __restrict__ part, int nb,
                                   float scale, float* __restrict__ C) {
  int idx = blockIdx.x * blockDim.x + threadIdx.x;
  if (idx >= 4096) return;
  float s = 0.f;
  for (int b = 0; b < nb; ++b) s += part[(size_t)b * 4096 + idx];
  C[idx] = s * scale;
}

// ---------------------------------------------------------------------------
// V_left / V_right (exact _V_lr_matrices)
// ---------------------------------------------------------------------------
__global__ void gramVlr_kernel(const float* __restrict__ V,
                               float* __restrict__ Vl, float* __restrict__ Vr) {
  int idx = blockIdx.x * blockDim.x + threadIdx.x;
  if (idx >= 64 * 64) return;
  int i = idx >> 6, j = idx & 63;
  float sl = 0.f, sr = 0.f;
  for (int k = 0; k < NVg; ++k) {
    float vik = V[(size_t)i * NVg + k];
    if (k >= 1)       sl += vik * (V[(size_t)j * NVg + k] - V[(size_t)j * NVg + k - 1]);
    if (k < NVg - 1)  sr += vik * (V[(size_t)j * NVg + k + 1] - V[(size_t)j * NVg + k]);
  }
  Vl[idx] = sl;
  Vr[idx] = sr;
}

// ---------------------------------------------------------------------------
// small helper kernels
// ---------------------------------------------------------------------------
__global__ void transpose64_kernel(const float* __restrict__ A, float* __restrict__ At) {
  int idx = blockIdx.x * blockDim.x + threadIdx.x;
  if (idx >= 4096) return;
  int i = idx >> 6, j = idx & 63;
  At[j * 64 + i] = A[i * 64 + j];
}

__global__ __launch_bounds__(128)
void ghost2_kernel(const float* __restrict__ M, const float* __restrict__ K,
                   int N, int c0, int c1, float* __restrict__ out) {
  int tid = threadIdx.x;
  int r = (tid < 64) ? tid : tid - 64;
  int c = (tid < 64) ? c0 : c1;
  float s = 0.f;
  for (int j = 0; j < 64; ++j) s += M[r * 64 + j] * K[(size_t)j * N + c];
  out[r * 2 + ((tid < 64) ? 0 : 1)] = s;
}

__global__ __launch_bounds__(64)
void ghost1_kernel(const float* __restrict__ M, const float* __restrict__ K,
                   int N, int c, float* __restrict__ out) {
  int r = threadIdx.x;
  float s = 0.f;
  for (int j = 0; j < 64; ++j) s += M[r * 64 + j] * K[(size_t)j * N + c];
  out[r] = s;
}

static __device__ inline float cell_fetch(const float* U, int n, int r, int j,
                                          const float* gL, const float* gR) {
  if (j < 2)      return gL ? gL[r * 2 + j] : 0.f;
  if (j < n + 2)  return U[(size_t)r * n + (j - 2)];
  return gR ? gR[r * 2 + (j - n - 2)] : 0.f;
}

__global__ void limiter_kernel(const float* __restrict__ U, int n,
                               const float* __restrict__ gL, const float* __restrict__ gR,
                               float* __restrict__ left, float* __restrict__ right) {
  int idx = blockIdx.x * blockDim.x + threadIdx.x;
  if (idx >= 64 * (n + 1)) return;
  int r = idx / (n + 1), i = idx % (n + 1);
  float c0 = cell_fetch(U, n, r, i,     gL, gR);
  float c1 = cell_fetch(U, n, r, i + 1, gL, gR);
  float c2 = cell_fetch(U, n, r, i + 2, gL, gR);
  float c3 = cell_fetch(U, n, r, i + 3, gL, gR);
  left [idx] = c1 + 0.5f * minmodf(c1 - c0, c2 - c1);
  right[idx] = c2 - 0.5f * minmodf(c2 - c1, c3 - c2);
}

__global__ void diff1_kernel(const float* __restrict__ K, int n,
                             const float* __restrict__ gl, const float* __restrict__ gr,
                             float* __restrict__ Kdl, float* __restrict__ Kdr,
                             float invdh) {
  int idx = blockIdx.x * blockDim.x + threadIdx.x;
  if (idx >= 64 * n) return;
  int r = idx / n, m = idx % n;
  auto cf = [&](int j) -> float {
    if (j == 0) return gl[r];
    if (j <= n) return K[(size_t)r * n + (j - 1)];
    return gr[r];
  };
  Kdl[idx] = (cf(m + 1) - cf(m))     * invdh;
  Kdr[idx] = (cf(m + 2) - cf(m + 1)) * invdh;
}

__global__ void scale_cols_pm_kernel(const float* __restrict__ K, const float* __restrict__ E,
                                     float* __restrict__ outP, float* __restrict__ outM, int n) {
  int idx = blockIdx.x * blockDim.x + threadIdx.x;
  if (idx >= 64 * n) return;
  int x = idx % n;
  float e = E[x], k = K[idx];
  outP[idx] = k * fmaxf(e, 0.f);
  outM[idx] = k * fminf(e, 0.f);
}

__global__ void rhs_combine_kernel(const float* __restrict__ F, const float* __restrict__ EF,
                                   float* __restrict__ out, int n, float invdh, float fac) {
  int idx = blockIdx.x * blockDim.x + threadIdx.x;
  if (idx >= 64 * n) return;
  int r = idx / n, i = idx % n;
  float d = (F[(size_t)r * (n + 1) + i + 1] - F[(size_t)r * (n + 1) + i]) * invdh;
  out[idx] = -d - fac * EF[idx];
}

__global__ void vs_combine_kernel(const float* __restrict__ A, const float* __restrict__ B,
                                  float* __restrict__ out) {
  int idx = blockIdx.x * blockDim.x + threadIdx.x;
  if (idx >= 64 * NVg) return;
  float v = vsf(idx % NVg);
  out[idx] = fmaxf(v, 0.f) * A[idx] + fminf(v, 0.f) * B[idx];
}

__global__ void scale_vec_kernel(const float* __restrict__ a, float* __restrict__ o,
                                 int n, float fac) {
  int i = blockIdx.x * blockDim.x + threadIdx.x;
  if (i < n) o[i] = fac * a[i];
}

__global__ void axpy_kernel(const float* __restrict__ a, const float* __restrict__ b,
                            float* __restrict__ o, int n, float dt) {
  int i = blockIdx.x * blockDim.x + threadIdx.x;
  if (i < n) o[i] = a[i] + dt * b[i];
}

__global__ void ssp2_kernel(const float* __restrict__ a, const float* __restrict__ b,
                            const float* __restrict__ c, float* __restrict__ o,
                            int n, float dt) {
  int i = blockIdx.x * blockDim.x + threadIdx.x;
  if (i < n) o[i] = 0.5f * (a[i] + b[i] + dt * c[i]);
}

__global__ __launch_bounds__(64)
void mass_kernel(const float* __restrict__ V, float* __restrict__ out) {
  int r = threadIdx.x;
  float s = 0.f;
  for (int k = 0; k < NVg; ++k) s += V[(size_t)r * NVg + k];
  out[r] = s * DVc;
}

__global__ __launch_bounds__(64)
void matvec64_kernel(const float* __restrict__ M, const float* __restrict__ v,
                     float* __restrict__ out) {
  int r = threadIdx.x;
  float s = 0.f;
  for (int j = 0; j < 64; ++j) s += M[r * 64 + j] * v[j];
  out[r] = s;
}

__global__ void gemvT_acc_kernel(const float* __restrict__ X, const float* __restrict__ coef,
                                 float* __restrict__ rho, int n, float Z, int init) {
  __shared__ float cs[64];
  if (threadIdx.x < 64) cs[threadIdx.x] = coef[threadIdx.x];
  __syncthreads();
  int x = blockIdx.x * blockDim.x + threadIdx.x;
  if (x >= n) return;
  float s = 0.f;
  for (int r = 0; r < 64; ++r) s += X[(size_t)r * n + x] * cs[r];
  rho[x] = (init ? 0.f : rho[x]) + Z * s;
}

// Poisson: E = antiderivative(rho) with zero mean (== FFT division by ik)
__global__ __launch_bounds__(1024)
void scanE_kernel(const float* __restrict__ rho, float* __restrict__ E) {
  __shared__ float ssum[1024];
  const int tid = threadIdx.x;
  float loc[16];
  float s = 0.f;
#pragma unroll
  for (int j = 0; j < 16; ++j) { loc[j] = rho[tid * 16 + j]; s += loc[j]; }
  ssum[tid] = s;
  __syncthreads();
  for (int off = 1; off < 1024; off <<= 1) {
    float v = (tid >= off) ? ssum[tid - off] : 0.f;
    __syncthreads();
    if (tid >= off) ssum[tid] += v;
    __syncthreads();
  }
  float base = ssum[tid] - s;
  __syncthreads();
  float P[16], p = base, psum = 0.f;
#pragma unroll
  for (int j = 0; j < 16; ++j) { p += loc[j]; P[j] = p * DXc; psum += P[j]; }
  ssum[tid] = psum;
  __syncthreads();
  for (int off = 512; off > 0; off >>= 1) {
    if (tid < off) ssum[tid] += ssum[tid + off];
    __syncthreads();
  }
  float mean = ssum[0] / (float)NXg;
#pragma unroll
  for (int j = 0; j < 16; ++j) E[tid * 16 + j] = P[j] - mean;
}

__global__ __launch_bounds__(64)
void chol64_kernel(const float* __restrict__ G, float* __restrict__ Lc,
                   float* __restrict__ Sup) {
  __shared__ float Gs[64][65];
  int tid = threadIdx.x;
  for (int i = tid; i < 4096; i += 64) Gs[i >> 6][i & 63] = G[i];
  __syncthreads();
  for (int k = 0; k < 64; ++k) {
    if (tid == k) Gs[k][k] = sqrtf(fmaxf(Gs[k][k], 1e-30f));
    __syncthreads();
    float dk = Gs[k][k];
    if (tid > k) Gs[tid][k] /= dk;
    __syncthreads();
    if (tid > k) {
      float lik = Gs[tid][k];
      for (int j = k + 1; j <= tid; ++j) Gs[tid][j] -= lik * Gs[j][k];
    }
    __syncthreads();
  }
  for (int i = tid; i < 4096; i += 64) {
    int r = i >> 6, c = i & 63;
    Lc [i] = (c <= r) ? Gs[r][c] : 0.f;
    Sup[i] = (c >= r) ? Gs[c][r] : 0.f;
  }
}

__global__ __launch_bounds__(64)
void trsm_kernel(const float* __restrict__ Lc, const float* __restrict__ Kin,
                 float* __restrict__ Xout, int n) {
  __shared__ float Ls[64][65];
  int tid = threadIdx.x;
  for (int i = tid; i < 4096; i += 64) Ls[i >> 6][i & 63] = Lc[i];
  __syncthreads();
  int col = blockIdx.x * 64 + tid;
  if (col >= n) return;
  float y[64];
#pragma unroll
  for (int i = 0; i < 64; ++i) {
    float s = Kin[(size_t)i * n + col];
    for (int j = 0; j < i; ++j) s -= Ls[i][j] * y[j];
    y[i] = s / Ls[i][i];
  }
#pragma unroll
  for (int i = 0; i < 64; ++i) Xout[(size_t)i * n + col] = y[i];
}

// ---------------------------------------------------------------------------
// Host orchestration
// ---------------------------------------------------------------------------
namespace {

struct Ctx {
  float *s0, *s1, *s2, *F, *left, *right, *s6, *s7, *s8, *Kq, *Kdl, *Kdr;
  float *rho, *E, *Efac, *Gpart;
  float *v_plus, *v_minus, *VmW, *VpW, *Vl, *Vr, *VlT, *VrT, *St, *G, *Lc, *Sq,
        *EpM, *EmM, *KL, *KR, *T1, *S1, *SR, *VmW2, *VpW2;
  float *gL, *gR, *gl1, *gr1, *massv, *coef;
  hipStream_t st;
};

inline void gram64(const float* A, const float* B, int Kd, float* C, int mode,
                   const float* wv, float scale, Ctx& c) {
  int nb = Kd >> 10;
  if (nb < 1) nb = 1;
  gram64_kernel<<<nb, 512, 0, c.st>>>(A, B, Kd, c.Gpart, mode, wv);
  gram_reduce_kernel<<<16, 256, 0, c.st>>>(c.Gpart, nb, scale, C);
}

inline void species_matrices(const float* V, Ctx& c) {
  gram64(V, V, NVg, c.v_plus, 1, nullptr, DVc, c);
  gram64(V, V, NVg, c.v_minus, 2, nullptr, DVc, c);
  gram64(V, V, NVg, c.VmW, 3, nullptr, DVc, c);
  gram64(V, V, NVg, c.VpW, 4, nullptr, DVc, c);
  gramVlr_kernel<<<16, 256, 0, c.st>>>(V, c.Vl, c.Vr);
  transpose64_kernel<<<16, 256, 0, c.st>>>(c.Vl, c.VlT);
  transpose64_kernel<<<16, 256, 0, c.st>>>(c.Vr, c.VrT);
}

inline void poisson6(const float* Xe, const float* Se, const float* Ve,
                     const float* Xi, const float* Si, const float* Vi, Ctx& c) {
  mass_kernel<<<1, 64, 0, c.st>>>(Ve, c.massv);
  matvec64_kernel<<<1, 64, 0, c.st>>>(Se, c.massv, c.coef);
  gemvT_acc_kernel<<<CDIV(NXg, 256), 256, 0, c.st>>>(Xe, c.coef, c.rho, NXg, -1.f, 1);
  mass_kernel<<<1, 64, 0, c.st>>>(Vi, c.massv);
  matvec64_kernel<<<1, 64, 0, c.st>>>(Si, c.massv, c.coef);
  gemvT_acc_kernel<<<CDIV(NXg, 256), 256, 0, c.st>>>(Xi, c.coef, c.rho, NXg, 1.f, 0);
  scanE_kernel<<<1, 1024, 0, c.st>>>(c.rho, c.E);
}

inline void k_rhs(const float* K, float fac, float* out, Ctx& c) {
  const int n = NXg;
  ghost2_kernel<<<1, 128, 0, c.st>>>(c.VmW, K, n, 1, 0, c.gL);
  ghost2_kernel<<<1, 128, 0, c.st>>>(c.VpW, K, n, n - 1, n - 2, c.gR);
  limiter_kernel<<<CDIV(64 * (n + 1), 256), 256, 0, c.st>>>(K, n, c.gL, c.gR, c.left, c.right);
  gemm64_kernel<<<CDIV(n + 1, 64), 128, 0, c.st>>>(c.v_plus, c.left, c.F, n + 1, 1.f, 0.f);
  gemm64_kernel<<<CDIV(n + 1, 64), 128, 0, c.st>>>(c.v_minus, c.right, c.F, n + 1, 1.f, 1.f);
  scale_cols_pm_kernel<<<CDIV(64 * n, 256), 256, 0, c.st>>>(K, c.E, c.s6, c.s7, n);
  gemm64_kernel<<<CDIV(n, 64), 128, 0, c.st>>>(c.Vl, c.s6, c.s8, n, 1.f, 0.f);
  gemm64_kernel<<<CDIV(n, 64), 128, 0, c.st>>>(c.Vr, c.s7, c.s8, n, 1.f, 1.f);
  rhs_combine_kernel<<<CDIV(64 * n, 256), 256, 0, c.st>>>(c.F, c.s8, out, n, 1.f / DXc, fac);
}

inline void klr_from(const float* S, const float* X, const float* VmM,
                     const float* VpM, Ctx& c) {
  const int n = NXg;
  transpose64_kernel<<<16, 256, 0, c.st>>>(S, c.St);
  gemm64_kernel<<<CDIV(n, 64), 128, 0, c.st>>>(c.St, X, c.Kq, n, 1.f, 0.f);
  ghost1_kernel<<<1, 64, 0, c.st>>>(VmM, c.Kq, n, 0, c.gl1);
  ghost1_kernel<<<1, 64, 0, c.st>>>(VpM, c.Kq, n, n - 1, c.gr1);
  diff1_kernel<<<CDIV(64 * n, 256), 256, 0, c.st>>>(c.Kq, n, c.gl1, c.gr1, c.Kdl, c.Kdr, 1.f / DXc);
  gram64(X, c.Kdl, n, c.KL, 0, nullptr, DXc, c);
  gram64(X, c.Kdr, n, c.KR, 0, nullptr, DXc, c);
}

inline void s_rhs(const float* S, const float* X, float fac, float* out, Ctx& c) {
  klr_from(S, X, c.VmW, c.VpW, c);
  gemm64_kernel<<<1, 128, 0, c.st>>>(c.KL, c.v_plus, out, 64, 1.f, 0.f);
  gemm64_kernel<<<1, 128, 0, c.st>>>(c.KR, c.v_minus, out, 64, 1.f, 1.f);
  gram64(X, X, NXg, c.EpM, 5, c.E, DXc, c);
  gram64(X, X, NXg, c.EmM, 6, c.E, DXc, c);
  gemm64_kernel<<<1, 128, 0, c.st>>>(c.EpM, S, c.T1, 64, 1.f, 0.f);
  gemm64_kernel<<<1, 128, 0, c.st>>>(c.T1, c.VlT, out, 64, fac, 1.f);
  gemm64_kernel<<<1, 128, 0, c.st>>>(c.EmM, S, c.T1, 64, 1.f, 0.f);
  gemm64_kernel<<<1, 128, 0, c.st>>>(c.T1, c.VrT, out, 64, fac, 1.f);
}

inline void cholqr(const float* M, int n, float scale, float* Sout, float* Qout, Ctx& c) {
  gram64(M, M, n, c.G, 0, nullptr, scale, c);
  chol64_kernel<<<1, 64, 0, c.st>>>(c.G, c.Lc, Sout);
  trsm_kernel<<<CDIV(n, 64), 64, 0, c.st>>>(c.Lc, M, Qout, n);
}

inline void l_rhs(const float* L, const float* X, float fac, float* out, Ctx& c) {
  const int nv = NVg;
  cholqr(L, nv, DVc, c.Sq, c.s6 /*Vq*/, c);
  gram64(c.s6, c.s6, nv, c.VmW2, 3, nullptr, DVc, c);
  gram64(c.s6, c.s6, nv, c.VpW2, 4, nullptr, DVc, c);
  klr_from(c.Sq, X, c.VmW2, c.VpW2, c);
  gemm64_kernel<<<CDIV(nv, 64), 128, 0, c.st>>>(c.KL, c.s6, c.s7 /*Tlv*/, nv, 1.f, 0.f);
  gemm64_kernel<<<CDIV(nv, 64), 128, 0, c.st>>>(c.KR, c.s6, c.s8 /*Trv*/, nv, 1.f, 0.f);
  vs_combine_kernel<<<CDIV(64 * nv, 256), 256, 0, c.st>>>(c.s7, c.s8, c.Kdl /*vf*/);
  scale_vec_kernel<<<CDIV(NXg, 256), 256, 0, c.st>>>(c.E, c.Efac, NXg, fac);
  gram64(X, X, NXg, c.EpM, 5, c.Efac, DXc, c);
  gram64(X, X, NXg, c.EmM, 6, c.Efac, DXc, c);
  limiter_kernel<<<CDIV(64 * (nv + 1), 256), 256, 0, c.st>>>(L, nv, nullptr, nullptr, c.left, c.right);
  gemm64_kernel<<<CDIV(nv + 1, 64), 128, 0, c.st>>>(c.EpM, c.left, c.F, nv + 1, 1.f, 0.f);
  gemm64_kernel<<<CDIV(nv + 1, 64), 128, 0, c.st>>>(c.EmM, c.right, c.F, nv + 1, 1.f, 1.f);
  rhs_combine_kernel<<<CDIV(64 * nv, 256), 256, 0, c.st>>>(c.F, c.Kdl, out, nv, 1.f / DVc, 1.f);
}

}  // namespace

extern "C" void kernel_launch(void* const* d_in, const int* in_sizes, int n_in,
                              void* d_out, int out_size, void* d_ws, size_t ws_size,
                              hipStream_t stream) {
  const float* Xe = (const float*)d_in[0];
  const float* Se = (const float*)d_in[1];
  const float* Ve = (const float*)d_in[2];
  const float* Xi = (const float*)d_in[3];
  const float* Si = (const float*)d_in[4];
  const float* Vi = (const float*)d_in[5];

  float* out = (float*)d_out;
  float* oXe = out;
  float* oSe = oXe + (size_t)R64 * NXg;
  float* oVe = oSe + (size_t)R64 * R64;
  float* oXi = oVe + (size_t)R64 * NVg;
  float* oSi = oXi + (size_t)R64 * NXg;
  float* oVi = oSi + (size_t)R64 * R64;

  Ctx c{};
  c.st = stream;
  float* ws = (float*)d_ws;
  const size_t BIG = (size_t)R64 * (NXg + 4);
  float* slot[12];
  for (int i = 0; i < 12; ++i) slot[i] = ws + (size_t)i * BIG;
  c.s0 = slot[0]; c.s1 = slot[1]; c.s2 = slot[2]; c.F = slot[3];
  c.left = slot[4]; c.right = slot[5]; c.s6 = slot[6]; c.s7 = slot[7];
  c.s8 = slot[8]; c.Kq = slot[9]; c.Kdl = slot[10]; c.Kdr = slot[11];
  float* sm = ws + 12 * BIG;
  c.rho = sm;  sm += NXg;
  c.E   = sm;  sm += NXg;
  c.Efac = sm; sm += NXg;
  c.Gpart = sm; sm += 16 * 4096;
  float** mats[] = {&c.v_plus, &c.v_minus, &c.VmW, &c.VpW, &c.Vl, &c.Vr,
                    &c.VlT, &c.VrT, &c.St, &c.G, &c.Lc, &c.Sq, &c.EpM, &c.EmM,
                    &c.KL, &c.KR, &c.T1, &c.S1, &c.SR, &c.VmW2, &c.VpW2};
  for (auto m : mats) { *m = sm; sm += 4096; }
  c.gL = sm; sm += 128;
  c.gR = sm; sm += 128;
  c.gl1 = sm; sm += 64;
  c.gr1 = sm; sm += 64;
  c.massv = sm; sm += 64;
  c.coef = sm; sm += 64;

  const float facs[2] = {-1.0f, 0.01f};
  const float* Xs[2] = {Xe, Xi};
  const float* Ss[2] = {Se, Si};
  const float* Vs[2] = {Ve, Vi};
  float* oX[2] = {oXe, oXi};
  float* oS[2] = {oSe, oSi};
  float* oV[2] = {oVe, oVi};
  const size_t NK = (size_t)R64 * NXg;
  const size_t NL = (size_t)R64 * NVg;

  // ---------------- stage 1: E from inputs, K-step ----------------
  poisson6(Xe, Se, Ve, Xi, Si, Vi, c);
  for (int sp = 0; sp < 2; ++sp) {
    species_matrices(Vs[sp], c);
    transpose64_kernel<<<16, 256, 0, stream>>>(Ss[sp], c.St);
    gemm64_kernel<<<CDIV(NXg, 64), 128, 0, stream>>>(c.St, Xs[sp], c.s0, NXg, 1.f, 0.f);
    k_rhs(c.s0, facs[sp], c.s2, c);
    axpy_kernel<<<CDIV((int)NK, 256), 256, 0, stream>>>(c.s0, c.s2, c.s1, (int)NK, DTc);
    k_rhs(c.s1, facs[sp], c.s2, c);
    ssp2_kernel<<<CDIV((int)NK, 256), 256, 0, stream>>>(c.s0, c.s1, c.s2, c.s0, (int)NK, DTc);
    cholqr(c.s0, NXg, DXc, oS[sp], oX[sp], c);
  }

  // ---------------- stage 2: E, S-step ----------------
  poisson6(oXe, oSe, Ve, oXi, oSi, Vi, c);
  for (int sp = 0; sp < 2; ++sp) {
    species_matrices(Vs[sp], c);
    s_rhs(oS[sp], oX[sp], facs[sp], c.SR, c);
    axpy_kernel<<<16, 256, 0, stream>>>(oS[sp], c.SR, c.S1, 4096, DTc);
    s_rhs(c.S1, oX[sp], facs[sp], c.SR, c);
    ssp2_kernel<<<16, 256, 0, stream>>>(oS[sp], c.S1, c.SR, oS[sp], 4096, DTc);
  }

  // ---------------- stage 3: E, L-step ----------------
  poisson6(oXe, oSe, Ve, oXi, oSi, Vi, c);
  for (int sp = 0; sp < 2; ++sp) {
    gemm64_kernel<<<CDIV(NVg, 64), 128, 0, stream>>>(oS[sp], Vs[sp], c.s0, NVg, 1.f, 0.f);
    l_rhs(c.s0, oX[sp], facs[sp], c.s2, c);
    axpy_kernel<<<CDIV((int)NL, 256), 256, 0, stream>>>(c.s0, c.s2, c.s1, (int)NL, DTc);
    l_rhs(c.s1, oX[sp], facs[sp], c.s2, c);
    ssp2_kernel<<<CDIV((int)NL, 256), 256, 0, stream>>>(c.s0, c.s1, c.s2, c.s0, (int)NL, DTc);
    cholqr(c.s0, NVg, DVc, oS[sp], oV[sp], c);
  }
}